// JKNet_10694468567473
// MI455X (gfx1250) — compile-verified
//
#include <hip/hip_runtime.h>

typedef float v2f __attribute__((ext_vector_type(2)));
typedef float v8f __attribute__((ext_vector_type(8)));

#define N_NODES 50000
#define E_EDGES 800000
#define C_IN    128
#define C_HID   96
#define C_OUT   64

#define ROW_TILES (N_NODES / 16)     // 3125, exact
#define WAVES_PER_BLOCK 8            // 256 threads

// ---------------------------------------------------------------------------
// WMMA fp32 GEMM:  Y[N x MOUT] = X[N x K] @ W[K x MOUT] (+ bias)
// W staged in LDS once per block (<= 48KB of 320KB WGP LDS).
// Each wave32 computes one 16-row strip x full MOUT via V_WMMA_F32_16X16X4_F32:
// A fragment from global (b64), B fragments from LDS, COLT accumulators.
// ---------------------------------------------------------------------------
template <int K, int MOUT>
__global__ void wmma_gemm_f32(const float* __restrict__ X,
                              const float* __restrict__ W,
                              const float* __restrict__ bias,
                              float* __restrict__ Y) {
    constexpr int COLT = MOUT / 16;
    __shared__ float Wlds[K * MOUT];

    for (int idx = threadIdx.x; idx < K * MOUT; idx += blockDim.x)
        Wlds[idx] = W[idx];
    __syncthreads();

    const int wave = threadIdx.x >> 5;
    const int lane = threadIdx.x & 31;
    const int rt = blockIdx.x * WAVES_PER_BLOCK + wave;
    if (rt >= ROW_TILES) return;       // wave-uniform -> EXEC all 1s for WMMA

    const int row0 = rt << 4;
    const int l16  = lane & 15;
    const int hi   = lane >> 4;        // 0: K pair {0,1}; 1: K pair {2,3}

    // A fragment: row = row0 + l16, cols k + 2*hi, k + 2*hi + 1 (ISA 16x4 layout)
    const float* __restrict__ xrow = X + (size_t)(row0 + l16) * K + (hi << 1);

    v8f acc[COLT];
#pragma unroll
    for (int t = 0; t < COLT; ++t) acc[t] = (v8f){};

    for (int k = 0; k < K; k += 4) {
        v2f a;
        a.x = xrow[k];
        a.y = xrow[k + 1];
        const float* wrow = Wlds + (k + (hi << 1)) * MOUT + l16;
#pragma unroll
        for (int t = 0; t < COLT; ++t) {
            v2f b;
            b.x = wrow[t * 16];            // row k+2*hi,   col t*16+l16
            b.y = wrow[t * 16 + MOUT];     // row k+2*hi+1, col t*16+l16
            acc[t] = __builtin_amdgcn_wmma_f32_16x16x4_f32(
                false, a, false, b, (short)0, acc[t], false, false);
        }
    }

    // C/D layout: VGPR v -> (M = v + 8*hi, N = l16)
    const int rbase = row0 + (hi << 3);
#pragma unroll
    for (int t = 0; t < COLT; ++t) {
        const int col = t * 16 + l16;
        const float badd = bias ? bias[col] : 0.0f;
#pragma unroll
        for (int v = 0; v < 8; ++v)
            Y[(size_t)(rbase + v) * MOUT + col] = acc[t][v] + badd;
    }
}

// ---------------------------------------------------------------------------
// Degree / symmetric normalization
// ---------------------------------------------------------------------------
__global__ void k_init_deg(float* __restrict__ deg) {
    int i = blockIdx.x * blockDim.x + threadIdx.x;
    if (i < N_NODES) deg[i] = 1.0f;            // self-loop
}

__global__ void k_count_deg(const int* __restrict__ dst, float* __restrict__ deg) {
    int e = blockIdx.x * blockDim.x + threadIdx.x;
    if (e < E_EDGES) atomicAdd(&deg[dst[e]], 1.0f);
}

__global__ void k_rsqrt(float* __restrict__ d) {
    int i = blockIdx.x * blockDim.x + threadIdx.x;
    if (i < N_NODES) d[i] = rsqrtf(d[i]);
}

// ---------------------------------------------------------------------------
// GCN aggregation: agg = scatter_add(Hw[src]*dis[src]*dis[dst]) + Hw*dis^2
// ---------------------------------------------------------------------------
__global__ void k_init_agg(const float* __restrict__ Hw,
                           const float* __restrict__ dis,
                           float* __restrict__ agg) {
    int idx = blockIdx.x * blockDim.x + threadIdx.x;
    if (idx >= N_NODES * C_HID) return;
    int i = idx / C_HID;
    float d = dis[i];
    agg[idx] = Hw[idx] * d * d;                // self-loop term
}

// One wave32 per edge; each lane covers 3 channels (96/32). src/dst/norm are
// wave-uniform; gathers are coalesced 32-wide.
__global__ void k_edge_scatter(const int* __restrict__ src,
                               const int* __restrict__ dst,
                               const float* __restrict__ dis,
                               const float* __restrict__ Hw,
                               float* __restrict__ agg) {
    const int wave = threadIdx.x >> 5;
    const int lane = threadIdx.x & 31;
    const int e = blockIdx.x * WAVES_PER_BLOCK + wave;
    if (e >= E_EDGES) return;
    const int s = src[e];
    const int d = dst[e];
    const float w = dis[s] * dis[d];
    const float* __restrict__ hs = Hw  + (size_t)s * C_HID;
    float* __restrict__       ad = agg + (size_t)d * C_HID;
    atomicAdd(&ad[lane],      hs[lane]      * w);
    atomicAdd(&ad[lane + 32], hs[lane + 32] * w);
    atomicAdd(&ad[lane + 64], hs[lane + 64] * w);
}

__global__ void k_post(const float* __restrict__ agg,
                       const float* __restrict__ bias,
                       float* __restrict__ hcur,
                       float* __restrict__ hmax) {
    int idx = blockIdx.x * blockDim.x + threadIdx.x;
    if (idx >= N_NODES * C_HID) return;
    int c = idx % C_HID;
    float v = agg[idx] + bias[c];
    v = fmaxf(v, 0.0f);                        // relu
    hcur[idx] = v;
    hmax[idx] = fmaxf(hmax[idx], v);           // JumpingKnowledge 'max'
}

// ---------------------------------------------------------------------------
extern "C" void kernel_launch(void* const* d_in, const int* in_sizes, int n_in,
                              void* d_out, int out_size, void* d_ws, size_t ws_size,
                              hipStream_t stream) {
    const float* x   = (const float*)d_in[0];
    const int*   ei  = (const int*)d_in[1];
    const int*   src = ei;
    const int*   dst = ei + E_EDGES;
    const float* Wp  = (const float*)d_in[2];
    const float* W0  = (const float*)d_in[3];
    const float* b0  = (const float*)d_in[4];
    const float* W1  = (const float*)d_in[5];
    const float* b1  = (const float*)d_in[6];
    const float* W2  = (const float*)d_in[7];
    const float* b2  = (const float*)d_in[8];
    const float* Wc  = (const float*)d_in[9];
    const float* bc  = (const float*)d_in[10];
    float* out = (float*)d_out;

    // Workspace layout (floats)
    float* ws   = (float*)d_ws;
    float* dis  = ws;                                   // N
    float* hmax = dis  + N_NODES;                       // N*96
    float* hcur = hmax + (size_t)N_NODES * C_HID;       // N*96
    float* Hw   = hcur + (size_t)N_NODES * C_HID;       // N*96
    float* agg  = Hw   + (size_t)N_NODES * C_HID;       // N*96

    const int GRID_GEMM = (ROW_TILES + WAVES_PER_BLOCK - 1) / WAVES_PER_BLOCK; // 391
    const int GRID_EDGE = (E_EDGES + WAVES_PER_BLOCK - 1) / WAVES_PER_BLOCK;   // 100000
    const int NC        = N_NODES * C_HID;
    const int GRID_NC   = (NC + 255) / 256;                                    // 18750

    // --- degree + symmetric norm ---
    k_init_deg <<<(N_NODES + 255) / 256, 256, 0, stream>>>(dis);
    k_count_deg<<<(E_EDGES + 255) / 256, 256, 0, stream>>>(dst, dis);
    k_rsqrt    <<<(N_NODES + 255) / 256, 256, 0, stream>>>(dis);

    // --- reps[0] = x @ Wp -> initial JK max ---
    wmma_gemm_f32<C_IN, C_HID><<<GRID_GEMM, 256, 0, stream>>>(x, Wp, nullptr, hmax);

    // --- layer 0 (K=128) ---
    wmma_gemm_f32<C_IN, C_HID><<<GRID_GEMM, 256, 0, stream>>>(x, W0, nullptr, Hw);
    k_init_agg    <<<GRID_NC,   256, 0, stream>>>(Hw, dis, agg);
    k_edge_scatter<<<GRID_EDGE, 256, 0, stream>>>(src, dst, dis, Hw, agg);
    k_post        <<<GRID_NC,   256, 0, stream>>>(agg, b0, hcur, hmax);

    // --- layer 1 (K=96) ---
    wmma_gemm_f32<C_HID, C_HID><<<GRID_GEMM, 256, 0, stream>>>(hcur, W1, nullptr, Hw);
    k_init_agg    <<<GRID_NC,   256, 0, stream>>>(Hw, dis, agg);
    k_edge_scatter<<<GRID_EDGE, 256, 0, stream>>>(src, dst, dis, Hw, agg);
    k_post        <<<GRID_NC,   256, 0, stream>>>(agg, b1, hcur, hmax);

    // --- layer 2 (K=96) ---
    wmma_gemm_f32<C_HID, C_HID><<<GRID_GEMM, 256, 0, stream>>>(hcur, W2, nullptr, Hw);
    k_init_agg    <<<GRID_NC,   256, 0, stream>>>(Hw, dis, agg);
    k_edge_scatter<<<GRID_EDGE, 256, 0, stream>>>(src, dst, dis, Hw, agg);
    k_post        <<<GRID_NC,   256, 0, stream>>>(agg, b2, hcur, hmax);

    // --- classifier: out = hmax @ Wc + bc (K=96, MOUT=64) ---
    wmma_gemm_f32<C_HID, C_OUT><<<GRID_GEMM, 256, 0, stream>>>(hmax, Wc, bc, out);
}